// NumberTheoryTPRModel_85126251807098
// MI455X (gfx1250) — compile-verified
//
#include <hip/hip_runtime.h>
#include <stdint.h>

#define GCD_ITERS 48

// Exact integer mod for integer-valued fp32 x in [0, 2^20), y in [1, 2^20).
// t = fma(x, rcp(y), -0.25) satisfies t - x/y in [-0.4, -0.1], so
// floor(t) is q_true or q_true-1; one conditional subtract fixes r exactly.
// All intermediates are exact integers (< 2^21) via FMA.
__device__ __forceinline__ float fast_imod(float x, float y) {
    float q = floorf(__builtin_fmaf(x, __builtin_amdgcn_rcpf(y), -0.25f));
    float r = __builtin_fmaf(-q, y, x);
    return (r >= y) ? (r - y) : r;
}

// One Euclid step. y is always a non-negative integer, so fmax(y,1) is the
// divide-by-zero guard (1 op). When y==0: r = x mod 1 = 0, so y stays 0
// without an extra select; x is held via one cndmask.
#define GS(x, y) do {                                                        \
    float ys_ = fmaxf((y), 1.0f);                                            \
    float r_  = fast_imod((x), ys_);                                         \
    (x) = ((y) != 0.0f) ? (y) : (x);                                         \
    (y) = r_;                                                                \
} while (0)

__device__ __forceinline__ float scalar_result(float a, float b, int op,
                                               float wm, float wg) {
    float m = fast_imod(a, b);
    float x = a, y = b;
    #pragma unroll 1
    for (int it = 0; it < GCD_ITERS; ++it) {
        if (y == 0.0f) break;
        GS(x, y);
    }
    return (op == 0) ? m * wm : ((op == 1) ? x * wg : 0.0f);
}

__global__ __launch_bounds__(256)
void nt_tpr_kernel(const float* __restrict__ a, const float* __restrict__ b,
                   const int* __restrict__ op,
                   const float* __restrict__ Wm, const float* __restrict__ Wg,
                   float* __restrict__ out, int n) {
    __shared__ float s_w[2];

#if defined(__gfx1250__)
    // Broadcast the two scalar weights through LDS via the gfx1250 async
    // global->LDS engine (one lane issues; ASYNCcnt tracks completion).
    if (threadIdx.x == 0) {
        unsigned lds0 = (unsigned)(uintptr_t)&s_w[0];
        unsigned lds1 = (unsigned)(uintptr_t)&s_w[1];
        uint64_t ga0 = (uint64_t)(uintptr_t)Wm;
        uint64_t ga1 = (uint64_t)(uintptr_t)Wg;
        asm volatile(
            "global_load_async_to_lds_b32 %0, %2, off\n\t"
            "global_load_async_to_lds_b32 %1, %3, off"
            :: "v"(lds0), "v"(lds1), "v"(ga0), "v"(ga1)
            : "memory");
    }
    asm volatile("s_wait_asynccnt 0" ::: "memory");
    __syncthreads();
    const float wm = s_w[0];
    const float wg = s_w[1];
#else
    const float wm = Wm[0];
    const float wg = Wg[0];
#endif

    const int n4     = n >> 2;
    const int stride = (int)(gridDim.x * blockDim.x);
    const int gtid   = (int)(blockIdx.x * blockDim.x + threadIdx.x);

    const float4* __restrict__ a4 = reinterpret_cast<const float4*>(a);
    const float4* __restrict__ b4 = reinterpret_cast<const float4*>(b);
    const int4*   __restrict__ o4 = reinterpret_cast<const int4*>(op);
    float4*       __restrict__ y4 = reinterpret_cast<float4*>(out);

    for (int i = gtid; i < n4; i += stride) {
        // Prefetch next grid-stride tile (global_prefetch_b8).
        int nxt = i + stride;
        if (nxt < n4) {
            __builtin_prefetch(&a4[nxt], 0, 3);
            __builtin_prefetch(&b4[nxt], 0, 3);
            __builtin_prefetch(&o4[nxt], 0, 3);
        }

        const float4 av = a4[i];
        const float4 bv = b4[i];
        const int4   ov = o4[i];

        // Mod path: one fast_imod per element.
        const float m0 = fast_imod(av.x, bv.x);
        const float m1 = fast_imod(av.y, bv.y);
        const float m2 = fast_imod(av.z, bv.z);
        const float m3 = fast_imod(av.w, bv.w);

        // Gcd chains: only op==1 lanes need gcd; start the others at y=0 so
        // they converge immediately and the ballot tracks fewer live chains.
        float x0 = av.x, y0 = (ov.x == 1) ? bv.x : 0.0f;
        float x1 = av.y, y1 = (ov.y == 1) ? bv.y : 0.0f;
        float x2 = av.z, y2 = (ov.z == 1) ? bv.z : 0.0f;
        float x3 = av.w, y3 = (ov.w == 1) ? bv.w : 0.0f;

        #pragma unroll 1
        for (int it = 0; it < GCD_ITERS; it += 2) {
            bool any = (y0 != 0.0f) | (y1 != 0.0f) |
                       (y2 != 0.0f) | (y3 != 0.0f);
            if (__builtin_amdgcn_ballot_w32(any) == 0u) break;
            GS(x0, y0); GS(x1, y1); GS(x2, y2); GS(x3, y3);
            GS(x0, y0); GS(x1, y1); GS(x2, y2); GS(x3, y3);
        }

        float4 r;
        r.x = (ov.x == 0) ? m0 * wm : ((ov.x == 1) ? x0 * wg : 0.0f);
        r.y = (ov.y == 0) ? m1 * wm : ((ov.y == 1) ? x1 * wg : 0.0f);
        r.z = (ov.z == 0) ? m2 * wm : ((ov.z == 1) ? x2 * wg : 0.0f);
        r.w = (ov.w == 0) ? m3 * wm : ((ov.w == 1) ? x3 * wg : 0.0f);
        y4[i] = r;
    }

    // Scalar tail (n not divisible by 4); no-op for B = 2^23.
    for (int i = (n4 << 2) + gtid; i < n; i += stride) {
        out[i] = scalar_result(a[i], b[i], op[i], wm, wg);
    }
}

extern "C" void kernel_launch(void* const* d_in, const int* in_sizes, int n_in,
                              void* d_out, int out_size, void* d_ws, size_t ws_size,
                              hipStream_t stream) {
    const float* a  = (const float*)d_in[0];
    const float* b  = (const float*)d_in[1];
    const int*   op = (const int*)d_in[2];
    const float* Wm = (const float*)d_in[3];
    const float* Wg = (const float*)d_in[4];
    float* out = (float*)d_out;

    const int n  = in_sizes[0];
    const int n4 = n >> 2;

    const int threads = 256;               // 8 wave32 waves per workgroup
    int blocks = 2048;                     // 4 grid-stride tiles/thread at B=2^23
    const int needed = (n4 > 0 ? n4 : n);
    const int maxb = (needed + threads - 1) / threads;
    if (blocks > maxb) blocks = (maxb > 0 ? maxb : 1);

    nt_tpr_kernel<<<dim3(blocks), dim3(threads), 0, stream>>>(
        a, b, op, Wm, Wg, out, n);
}